// CustomCLIP_11407433138213
// MI455X (gfx1250) — compile-verified
//
#include <hip/hip_runtime.h>
#include <math.h>

// Problem constants (from reference setup_inputs)
constexpr int B    = 16384;
constexpr int D    = 1024;
constexpr int R    = 256;
constexpr int NDOM = 3;
constexpr int NTXT = 1380;

typedef float v2f __attribute__((ext_vector_type(2)));
typedef float v8f __attribute__((ext_vector_type(8)));

// -----------------------------------------------------------------------------
// Kernel 1: stable partition of row indices by domain label.
// One block of 256 threads; each thread owns a contiguous chunk of 64 rows.
// perm[pos] = original row index, rows grouped by domain, stable within domain.
// -----------------------------------------------------------------------------
__global__ __launch_bounds__(256) void partition_kernel(const int* __restrict__ labels,
                                                        int* __restrict__ perm) {
    __shared__ int cnt[256][3];
    __shared__ int base[256][3];
    __shared__ int tot[3];

    const int t = threadIdx.x;
    const int begin = t * 64;

    int c0 = 0, c1 = 0, c2 = 0;
    for (int j = 0; j < 64; ++j) {
        int l = labels[begin + j];
        c0 += (l == 0);
        c1 += (l == 1);
        c2 += (l == 2);
    }
    cnt[t][0] = c0; cnt[t][1] = c1; cnt[t][2] = c2;
    __syncthreads();

    if (t == 0) {
        int s0 = 0, s1 = 0, s2 = 0;
        for (int i = 0; i < 256; ++i) {
            base[i][0] = s0; s0 += cnt[i][0];
            base[i][1] = s1; s1 += cnt[i][1];
            base[i][2] = s2; s2 += cnt[i][2];
        }
        tot[0] = s0; tot[1] = s1; tot[2] = s2;
    }
    __syncthreads();

    int p0 = 0      + base[t][0];
    int p1 = tot[0] + base[t][1];
    int p2 = tot[0] + tot[1] + base[t][2];
    for (int j = 0; j < 64; ++j) {
        int row = begin + j;
        int l = labels[row];
        if (l == 0)      perm[p0++] = row;
        else if (l == 1) perm[p1++] = row;
        else             perm[p2++] = row;
    }
}

// -----------------------------------------------------------------------------
// Kernel 2: per 16-row tile (rows gathered through perm):
//   h = relu(x @ W1[d])           (16x1024 @ 1024x256)
//   a = relu(h @ W2[d])           (16x256  @ 256x1024)
//   fused = 0.2*a + 0.8*x ; fused /= ||fused||_2 (per row)
//   scatter fused back to ws at original row index.
// Block = 256 threads = 8 waves. WMMA f32 16x16x4 throughout.
// After partition, tiles are domain-uniform except <=2 boundary tiles; rows
// whose label != d are computed but masked at the fuse step (exact result).
// -----------------------------------------------------------------------------
__global__ __launch_bounds__(256) void adapter_kernel(const float* __restrict__ X,
                                                      const int*   __restrict__ labels,
                                                      const float* __restrict__ W1,
                                                      const float* __restrict__ W2,
                                                      const int*   __restrict__ perm,
                                                      float*       __restrict__ fusedOut) {
    __shared__ int   rowIdx[16];
    __shared__ int   rowLab[16];
    __shared__ float Hs[16][R + 4];        // stage-1 output (relu'd)
    __shared__ float Fus[16][D + 8];       // fused accumulator
    __shared__ float part[16][16];         // norm reduction

    const int tid  = threadIdx.x;
    const int wave = tid >> 5;
    const int lane = tid & 31;
    const int mrow = lane & 15;            // M index within tile for A/B frags
    const int kgrp = (lane >> 4) * 2;      // K sub-offset per ISA A-layout (0 or 2)

    const int p0 = blockIdx.x * 16;

    if (tid < 16) {
        int r = perm[p0 + tid];
        rowIdx[tid] = r;
        rowLab[tid] = labels[r];
    }
    __syncthreads();

    // Which domains appear in this tile (identical result in every thread).
    unsigned present = 0;
    for (int i = 0; i < 16; ++i) present |= 1u << rowLab[i];

    // Initialize fused accumulator with (1 - ratio) * x.
    for (int idx = tid; idx < 16 * D; idx += 256) {
        int rr = idx >> 10, cc = idx & (D - 1);
        Fus[rr][cc] = 0.8f * X[(size_t)rowIdx[rr] * D + cc];
    }

    const float* arowX = X + (size_t)rowIdx[mrow] * D;   // A-frag row pointer (stage 1)

    for (int d = 0; d < NDOM; ++d) {
        if (!((present >> d) & 1u)) continue;
        __syncthreads();   // Hs free to overwrite; Fus init done (first iter)

        // ---- Stage 1: H = relu(X_tile @ W1[d]) ----
        // Each wave owns two 16-col N-tiles sharing one A-fragment per K-step.
        const float* W1d = W1 + (size_t)d * D * R;
        {
            const int n0a = wave * 16;
            const int n0b = (wave + 8) * 16;
            v8f acc0 = {}, acc1 = {};
            const float* bb0 = W1d + n0a + mrow;
            const float* bb1 = W1d + n0b + mrow;
            for (int k = 0; k < D; k += 4) {
                v2f a, b0, b1;
                a.x = arowX[k + kgrp];
                a.y = arowX[k + kgrp + 1];
                const float* q0 = bb0 + (size_t)(k + kgrp) * R;
                const float* q1 = bb1 + (size_t)(k + kgrp) * R;
                b0.x = q0[0]; b0.y = q0[R];
                b1.x = q1[0]; b1.y = q1[R];
                acc0 = __builtin_amdgcn_wmma_f32_16x16x4_f32(
                    false, a, false, b0, (short)0, acc0, false, false);
                acc1 = __builtin_amdgcn_wmma_f32_16x16x4_f32(
                    false, a, false, b1, (short)0, acc1, false, false);
            }
            const int rbase = (lane >> 4) * 8;
            for (int v = 0; v < 8; ++v) {
                Hs[rbase + v][n0a + mrow] = fmaxf(acc0[v], 0.0f);
                Hs[rbase + v][n0b + mrow] = fmaxf(acc1[v], 0.0f);
            }
        }
        __syncthreads();

        // ---- Stage 2: A2 = relu(H @ W2[d]) ; fuse into Fus (masked by label) ----
        const float* W2d = W2 + (size_t)d * R * D;
        for (int j = 0; j < 8; ++j) {
            int n0 = (wave * 8 + j) * 16;
            v8f acc = {};
            const float* bbase = W2d + n0 + mrow;
            for (int k = 0; k < R; k += 4) {
                v2f a, b;
                a.x = Hs[mrow][k + kgrp];
                a.y = Hs[mrow][k + kgrp + 1];
                const float* bp = bbase + (size_t)(k + kgrp) * D;
                b.x = bp[0];
                b.y = bp[D];
                acc = __builtin_amdgcn_wmma_f32_16x16x4_f32(
                    false, a, false, b, (short)0, acc, false, false);
            }
            const int rbase = (lane >> 4) * 8;
            for (int v = 0; v < 8; ++v) {
                int rr = rbase + v;
                if (rowLab[rr] == d)
                    Fus[rr][n0 + mrow] += 0.2f * fmaxf(acc[v], 0.0f);
            }
        }
        __syncthreads();
    }
    __syncthreads();

    // ---- Row-wise L2 normalize + scatter store ----
    {
        int rr = tid >> 4, j = tid & 15;
        float s = 0.0f;
        for (int c = j; c < D; c += 16) {
            float v = Fus[rr][c];
            s += v * v;
        }
        part[rr][j] = s;
        __syncthreads();
        if (j == 0) {
            float t = 0.0f;
            for (int i = 0; i < 16; ++i) t += part[rr][i];
            part[rr][0] = 1.0f / sqrtf(t);
        }
        __syncthreads();
    }
    for (int idx = tid; idx < 16 * D; idx += 256) {
        int rr = idx >> 10, cc = idx & (D - 1);
        fusedOut[(size_t)rowIdx[rr] * D + cc] = Fus[rr][cc] * part[rr][0];
    }
}

// -----------------------------------------------------------------------------
// Kernel 3: logits = exp(scale) * fused @ text^T  (16384x1024 @ 1024x1380)
// Each wave owns FOUR 16x16 output tiles (16 rows x 64 cols) sharing a single
// A-fragment per K-step: 4 WMMAs per A-load -> 2.5 loads/wmma instead of 4.
// Grid: (ceil(1380/512)=3 col groups, 1024 row tiles), 256 threads = 8 waves.
// -----------------------------------------------------------------------------
__global__ __launch_bounds__(256) void logits_kernel(const float* __restrict__ fused,
                                                     const float* __restrict__ text,
                                                     const float* __restrict__ pscale,
                                                     float*       __restrict__ out) {
    const int tid  = threadIdx.x;
    const int wave = tid >> 5;
    const int lane = tid & 31;

    const int n0 = (blockIdx.x * 8 + wave) * 64;     // first of 4 col-tiles
    if (n0 >= NTXT) return;                          // wave-uniform exit

    const int m0   = blockIdx.y * 16;
    const int mrow = lane & 15;
    const int kgrp = (lane >> 4) * 2;

    const float* arow = fused + (size_t)(m0 + mrow) * D;

    const float* brow0;
    const float* brow1;
    const float* brow2;
    const float* brow3;
    {
        int c0 = n0 + 0  + mrow; if (c0 >= NTXT) c0 = NTXT - 1;
        int c1 = n0 + 16 + mrow; if (c1 >= NTXT) c1 = NTXT - 1;
        int c2 = n0 + 32 + mrow; if (c2 >= NTXT) c2 = NTXT - 1;
        int c3 = n0 + 48 + mrow; if (c3 >= NTXT) c3 = NTXT - 1;
        brow0 = text + (size_t)c0 * D;
        brow1 = text + (size_t)c1 * D;
        brow2 = text + (size_t)c2 * D;
        brow3 = text + (size_t)c3 * D;
    }

    v8f acc0 = {}, acc1 = {}, acc2 = {}, acc3 = {};
    for (int k = 0; k < D; k += 4) {
        const int kk = k + kgrp;
        v2f a, b0, b1, b2, b3;
        a.x  = arow[kk];   a.y  = arow[kk + 1];
        b0.x = brow0[kk];  b0.y = brow0[kk + 1];
        b1.x = brow1[kk];  b1.y = brow1[kk + 1];
        b2.x = brow2[kk];  b2.y = brow2[kk + 1];
        b3.x = brow3[kk];  b3.y = brow3[kk + 1];
        acc0 = __builtin_amdgcn_wmma_f32_16x16x4_f32(
            false, a, false, b0, (short)0, acc0, false, false);
        acc1 = __builtin_amdgcn_wmma_f32_16x16x4_f32(
            false, a, false, b1, (short)0, acc1, false, false);
        acc2 = __builtin_amdgcn_wmma_f32_16x16x4_f32(
            false, a, false, b2, (short)0, acc2, false, false);
        acc3 = __builtin_amdgcn_wmma_f32_16x16x4_f32(
            false, a, false, b3, (short)0, acc3, false, false);
    }

    const float s = expf(pscale[0]);
    const int rbase = m0 + (lane >> 4) * 8;

    {
        int cc = n0 + 0 + mrow;
        if (cc < NTXT)
            for (int v = 0; v < 8; ++v)
                out[(size_t)(rbase + v) * NTXT + cc] = s * acc0[v];
    }
    {
        int cc = n0 + 16 + mrow;
        if (cc < NTXT)
            for (int v = 0; v < 8; ++v)
                out[(size_t)(rbase + v) * NTXT + cc] = s * acc1[v];
    }
    {
        int cc = n0 + 32 + mrow;
        if (cc < NTXT)
            for (int v = 0; v < 8; ++v)
                out[(size_t)(rbase + v) * NTXT + cc] = s * acc2[v];
    }
    {
        int cc = n0 + 48 + mrow;
        if (cc < NTXT)
            for (int v = 0; v < 8; ++v)
                out[(size_t)(rbase + v) * NTXT + cc] = s * acc3[v];
    }
}

// -----------------------------------------------------------------------------
// Launch
// -----------------------------------------------------------------------------
extern "C" void kernel_launch(void* const* d_in, const int* in_sizes, int n_in,
                              void* d_out, int out_size, void* d_ws, size_t ws_size,
                              hipStream_t stream) {
    const float* X      = (const float*)d_in[0];   // image_features (B,D)
    const int*   labels = (const int*)  d_in[1];   // domain_labels (B,)
    const float* W1     = (const float*)d_in[2];   // (NDOM,D,R)
    const float* W2     = (const float*)d_in[3];   // (NDOM,R,D)
    const float* text   = (const float*)d_in[4];   // (NTXT,D), pre-normalized
    const float* scale  = (const float*)d_in[5];   // scalar
    float*       out    = (float*)d_out;           // (B,NTXT)

    // Workspace layout: perm (B ints, 64KB) | fused_hat (B*D f32, 64MB)
    int*   perm  = (int*)d_ws;
    float* fused = (float*)((char*)d_ws + (size_t)B * sizeof(int));

    partition_kernel<<<1, 256, 0, stream>>>(labels, perm);

    adapter_kernel<<<B / 16, 256, 0, stream>>>(X, labels, W1, W2, perm, fused);

    dim3 grid((NTXT + 64 * 8 - 1) / (64 * 8), B / 16);   // (3, 1024)
    logits_kernel<<<grid, 256, 0, stream>>>(fused, text, scale, out);
}